// CoPE_25391846654023
// MI455X (gfx1250) — compile-verified
//
#include <hip/hip_runtime.h>

// CoPE fused kernel for MI455X (gfx1250, wave32).
// Memory-bound: ~1.09 GB traffic -> ~47us ideal at 23.3 TB/s.
// LUT GEMM (65536x64x64 f32) done with v_wmma_f32_16x16x4_f32 (exact f32 path).

typedef float v2f __attribute__((ext_vector_type(2)));
typedef float v4f __attribute__((ext_vector_type(4)));
typedef float v8f __attribute__((ext_vector_type(8)));

#define S_LEN 2048
#define D_DIM 64
#define NPOSE 64
#define ROWS_PER_BLOCK 16

__global__ __launch_bounds__(512, 1)
void cope_kernel(const float* __restrict__ query,
                 const float* __restrict__ attn,
                 const float* __restrict__ pemb,
                 float* __restrict__ out)
{
    __shared__ float s_lut[ROWS_PER_BLOCK][NPOSE];  // stride 64 floats -> bank == index, conflict-free

    const int wg   = blockIdx.x;
    const int bh   = wg >> 7;           // wg / (S/16): which (b,h)
    const int s0   = (wg & 127) << 4;   // s-tile start
    const int tid  = threadIdx.x;
    const int wave = tid >> 5;
    const int lane = tid & 31;

    // ---------------- Phase 1: wave 0 builds lut[16][64] = Q[16x64] @ P[64x64]
    // via V_WMMA_F32_16X16X4_F32. Branch is wave-uniform -> EXEC all-1s inside.
    if (wave == 0) {
        const int hi = lane >> 4;   // half-wave select
        const int lo = lane & 15;
        const float* Qb = query + ((size_t)bh * S_LEN + s0) * D_DIM;

        v8f acc0 = {}; v8f acc1 = {}; v8f acc2 = {}; v8f acc3 = {};
        #pragma unroll
        for (int k = 0; k < 16; ++k) {
            const int d0 = 4 * k + 2 * hi;  // A VGPR0: K={k0,k0+2} per half-wave
            v2f a;
            a.x = Qb[lo * D_DIM + d0];
            a.y = Qb[lo * D_DIM + d0 + 1];
            v2f b0, b1, b2, b3;            // B VGPR0: K rows {k0,k0+2}, N striped on lanes
            b0.x = pemb[(d0    ) * NPOSE +      lo];
            b0.y = pemb[(d0 + 1) * NPOSE +      lo];
            b1.x = pemb[(d0    ) * NPOSE + 16 + lo];
            b1.y = pemb[(d0 + 1) * NPOSE + 16 + lo];
            b2.x = pemb[(d0    ) * NPOSE + 32 + lo];
            b2.y = pemb[(d0 + 1) * NPOSE + 32 + lo];
            b3.x = pemb[(d0    ) * NPOSE + 48 + lo];
            b3.y = pemb[(d0 + 1) * NPOSE + 48 + lo];
            acc0 = __builtin_amdgcn_wmma_f32_16x16x4_f32(false, a, false, b0, (short)0, acc0, false, false);
            acc1 = __builtin_amdgcn_wmma_f32_16x16x4_f32(false, a, false, b1, (short)0, acc1, false, false);
            acc2 = __builtin_amdgcn_wmma_f32_16x16x4_f32(false, a, false, b2, (short)0, acc2, false, false);
            acc3 = __builtin_amdgcn_wmma_f32_16x16x4_f32(false, a, false, b3, (short)0, acc3, false, false);
        }
        // C/D layout: VGPR j -> row M = j + 8*hi, col N = 16*t + lo
        #pragma unroll
        for (int j = 0; j < 8; ++j) {
            const int row = j + 8 * hi;
            s_lut[row][     lo] = acc0[j];
            s_lut[row][16 + lo] = acc1[j];
            s_lut[row][32 + lo] = acc2[j];
            s_lut[row][48 + lo] = acc3[j];
        }
    }
    __syncthreads();

    // ---------------- Phase 2: one wave per row (16 waves, 16 rows)
    const int srow = wave;
    const size_t rowbase = ((size_t)bh * S_LEN + (size_t)(s0 + srow)) * (size_t)S_LEN;
    const float* rp = attn + rowbase + (size_t)lane * 64;
    float*       op = out  + rowbase + (size_t)lane * 64;

    // Pass A: stream gates, keep per-element EXCLUSIVE local prefix in registers.
    // reverse-cumsum[i] = total - exclusive_prefix[i]  (gates never needed again)
    float e[64];
    float run = 0.0f;
    #pragma unroll
    for (int j4 = 0; j4 < 16; ++j4) {
        v4f x = __builtin_nontemporal_load((const v4f*)(rp + j4 * 4));
        #pragma unroll
        for (int c = 0; c < 4; ++c) {
            float g = __builtin_amdgcn_rcpf(1.0f + __expf(-x[c]));  // sigmoid
            e[j4 * 4 + c] = run;
            run += g;
        }
    }

    // Wave-level exclusive scan of lane totals (5 shuffle steps, wave32).
    float incl = run;
    #pragma unroll
    for (int d = 1; d < 32; d <<= 1) {
        float v = __shfl_up(incl, (unsigned)d, 32);
        if (lane >= d) incl += v;
    }
    const float total = __shfl(incl, 31, 32);
    const float base  = total - (incl - run);   // total - lane_exclusive_offset

    // Pass B: position, LUT lerp (1 sub + 1 FMA), NT store.
    const float* lut = &s_lut[srow][0];
    #pragma unroll
    for (int j4 = 0; j4 < 16; ++j4) {
        v4f o;
        #pragma unroll
        for (int c = 0; c < 4; ++c) {
            float pos = base - e[j4 * 4 + c];
            pos = fminf(pos, 63.0f);
            float pf = floorf(pos);
            float w  = pos - pf;
            int i0 = (int)pf;
            int i1 = min(i0 + 1, 63);
            float l0 = lut[i0];
            float l1 = lut[i1];
            o[c] = fmaf(w, l1 - l0, l0);   // == l1*w + l0*(1-w)
        }
        __builtin_nontemporal_store(o, (v4f*)(op + j4 * 4));
    }
}

extern "C" void kernel_launch(void* const* d_in, const int* in_sizes, int n_in,
                              void* d_out, int out_size, void* d_ws, size_t ws_size,
                              hipStream_t stream) {
    const float* query = (const float*)d_in[0];   // [2,16,2048,64]
    const float* attn  = (const float*)d_in[1];   // [2,16,2048,2048]
    const float* pemb  = (const float*)d_in[2];   // [1,64,64]
    float* out = (float*)d_out;                   // [2,16,2048,2048]

    // 2*16 (b,h) * 128 s-tiles of 16 rows = 4096 blocks, 512 threads (16 waves)
    dim3 grid(4096), block(512);
    hipLaunchKernelGGL(cope_kernel, grid, block, 0, stream, query, attn, pemb, out);
    (void)in_sizes; (void)n_in; (void)out_size; (void)d_ws; (void)ws_size;
}